// FGDLoss_14843406975340
// MI455X (gfx1250) — compile-verified
//
#include <hip/hip_runtime.h>
#include <stdint.h>

// Problem constants (match reference)
#define BQ 16
#define AQ 32768
#define GQ 100
#define NBQ 64                 // blocks per batch image for anchor-parallel kernels
#define ASLICE (AQ / NBQ)      // 512 anchors per block
#define NPART (BQ * NBQ)       // 1024 partials (power of two, 16 wmma chunks of 64)
#define OVERLAP_THRESH 0.5f

typedef float v8f __attribute__((ext_vector_type(8)));
typedef float v2f __attribute__((ext_vector_type(2)));

// ---------------------------------------------------------------------------
// Kernel 1: zero the per-GT argmax keys.
// key[b*G+g] = (float_bits(iou) << 32) | ~anchor  -> atomicMax picks max iou,
// then smallest anchor index (jnp.argmax first-wins tie-break). iou >= 0 for
// valid GTs so float bits are monotonic.
// ---------------------------------------------------------------------------
__global__ void fgd_init_kernel(unsigned long long* __restrict__ gkeys) {
    int i = blockIdx.x * blockDim.x + threadIdx.x;
    if (i < BQ * GQ) gkeys[i] = 0ull;
}

// ---------------------------------------------------------------------------
// Kernel 2: per-anchor best GT (max+argmax over G) and per-GT best anchor
// (via read-filtered global_atomic_max_u64 on packed keys).
// matchword: bit31 = (best_overlap >= 0.5), bits[30:0] = best GT index.
// ---------------------------------------------------------------------------
__global__ void fgd_match_kernel(const float* __restrict__ anchors,   // [B,A,2]
                                 const float* __restrict__ targets,   // [B,G,3]
                                 unsigned long long* __restrict__ gkeys,
                                 unsigned int* __restrict__ matchw) {
    const int b   = blockIdx.y;
    const int blk = blockIdx.x;
    const int tid = threadIdx.x;

    __shared__ float s_g0[GQ], s_g1[GQ], s_len[GQ];
    __shared__ int   s_gid[GQ];
    __shared__ int   s_nvalid;

    // Compact valid GTs in ascending g order (preserves argmax tie-break).
    if (tid == 0) {
        int n = 0;
        for (int g = 0; g < GQ; ++g) {
            const float* t = targets + ((size_t)b * GQ + g) * 3;
            float t0 = t[0], t1 = t[1], lab = t[2];
            if (lab > 0.0f) {
                s_g0[n] = t0; s_g1[n] = t1; s_len[n] = t1 - t0; s_gid[n] = g;
                ++n;
            }
        }
        s_nvalid = n;
    }
    __syncthreads();
    const int nv = s_nvalid;

    const int abase = blk * ASLICE;
    for (int i = tid; i < ASLICE; i += blockDim.x) {
        const int a   = abase + i;
        const int idx = b * AQ + a;
        const float a0 = anchors[(size_t)idx * 2 + 0];
        const float a1 = anchors[(size_t)idx * 2 + 1];
        const float la = a1 - a0;
        const unsigned int sec = ~(unsigned int)a;   // smaller anchor => larger key

        float bestv = -1.0f;
        int   bestg = 0;
        for (int j = 0; j < nv; ++j) {
            float inter = fminf(s_g1[j], a1) - fmaxf(s_g0[j], a0);
            inter = fmaxf(inter, 0.0f);
            float iou = inter / (s_len[j] + la - inter);
            int g = s_gid[j];
            if (iou > bestv) { bestv = iou; bestg = g; }   // strict > => first g wins
            unsigned long long key =
                ((unsigned long long)__float_as_uint(iou) << 32) | (unsigned long long)sec;
            unsigned long long* slot = &gkeys[b * GQ + g];
            // read-filter: value only grows, so a stale read can only cause a
            // redundant (never a missing) atomic
            if (key > *slot) atomicMax(slot, key);
        }
        matchw[idx] = (bestv >= OVERLAP_THRESH ? 0x80000000u : 0u) | (unsigned int)bestg;
    }
}

// ---------------------------------------------------------------------------
// Kernel 3: force-match scatter. Sequential over g per batch so duplicate
// best-anchor collisions resolve deterministically (later g wins).
// ---------------------------------------------------------------------------
__global__ void fgd_scatter_kernel(const float* __restrict__ targets,
                                   const unsigned long long* __restrict__ gkeys,
                                   unsigned int* __restrict__ matchw) {
    const int b = blockIdx.x;
    if (threadIdx.x != 0) return;
    for (int g = 0; g < GQ; ++g) {
        float lab = targets[((size_t)b * GQ + g) * 3 + 2];
        if (lab > 0.0f) {
            unsigned long long key = gkeys[b * GQ + g];
            unsigned int a = ~(unsigned int)(key & 0xffffffffull);
            matchw[b * AQ + (int)a] = 0x80000000u | (unsigned int)g;  // overlap := 2.0
        }
    }
}

// ---------------------------------------------------------------------------
// Kernel 4: smooth-L1 localization loss partials per block (deterministic
// LDS tree reduction; no float atomics).
// ---------------------------------------------------------------------------
__global__ void fgd_loss_kernel(const float* __restrict__ loc_pred,   // [B,A,2]
                                const float* __restrict__ anchors,    // [B,A,2]
                                const int*   __restrict__ ignore,     // [B,A]
                                const float* __restrict__ targets,    // [B,G,3]
                                const unsigned int* __restrict__ matchw,
                                float* __restrict__ pl, float* __restrict__ pc) {
    const int b   = blockIdx.y;
    const int blk = blockIdx.x;
    const int tid = threadIdx.x;

    __shared__ float s_b0[GQ], s_b1[GQ], s_lab[GQ];
    for (int g = tid; g < GQ; g += blockDim.x) {
        const float* t = targets + ((size_t)b * GQ + g) * 3;
        s_b0[g] = t[0]; s_b1[g] = t[1]; s_lab[g] = t[2];
    }
    __syncthreads();

    float lsum = 0.0f, csum = 0.0f;
    const int abase = b * AQ + blk * ASLICE;
    for (int i = tid; i < ASLICE; i += blockDim.x) {
        const int idx = abase + i;
        const unsigned int w = matchw[idx];
        if ((w >> 31) && ignore[idx] == 0) {
            const int g = (int)(w & 0x7fffffffu);
            if (s_lab[g] > 0.0f) {                     // positive anchor
                float a0 = anchors[(size_t)idx * 2 + 0];
                float a1 = anchors[(size_t)idx * 2 + 1];
                float ac  = (a0 + a1) * 0.5f;
                float asz = a1 - a0;
                float g0 = s_b0[g], g1 = s_b1[g];
                float gc  = (g0 + g1) * 0.5f;
                float gsz = g1 - g0;
                float t0 = (gc - ac) / (0.1f * asz);   // VAR0 = 0.1
                float t1 = logf(gsz / asz) * 5.0f;     // / VAR1 (0.2)
                float d0 = loc_pred[(size_t)idx * 2 + 0] - t0;
                float d1 = loc_pred[(size_t)idx * 2 + 1] - t1;
                float ad0 = fabsf(d0), ad1 = fabsf(d1);
                lsum += (ad0 < 1.0f ? 0.5f * d0 * d0 : ad0 - 0.5f);
                lsum += (ad1 < 1.0f ? 0.5f * d1 * d1 : ad1 - 0.5f);
                csum += 1.0f;
            }
        }
    }

    __shared__ float rl[256], rc[256];
    rl[tid] = lsum; rc[tid] = csum;
    __syncthreads();
    for (int s = 128; s > 0; s >>= 1) {
        if (tid < s) { rl[tid] += rl[tid + s]; rc[tid] += rc[tid + s]; }
        __syncthreads();
    }
    if (tid == 0) {
        pl[b * NBQ + blk] = rl[0];
        pc[b * NBQ + blk] = rc[0];
    }
}

// ---------------------------------------------------------------------------
// Kernel 5: finalize via V_WMMA_F32_16X16X4_F32 row-sum reduction.
// One wave (32 threads, EXEC all ones). B = all-ones => D[m,n] = sum_k A[m,k];
// accumulate C across 16 chunks of 64 partials, for loss and count separately.
// Column value is n-independent; total = (per-lane reg sum of lane L) +
// (lane L^16), combined with a wave32 shuffle.
// A-matrix 16x4 f32 layout: lanes 0-15 hold K=0,1; lanes 16-31 hold K=2,3.
// ---------------------------------------------------------------------------
__global__ void fgd_finalize_kernel(const float* __restrict__ pl,
                                    const float* __restrict__ pc,
                                    float* __restrict__ out) {
    const int lane = threadIdx.x;            // 0..31, one full wave
    const int m    = lane & 15;
    const int koff = (lane >> 4) * 2;

    __builtin_amdgcn_s_wait_tensorcnt(0);    // CDNA5 split-counter wait (no-op here)

    v2f ones; ones.x = 1.0f; ones.y = 1.0f;
    v8f cl = {}; v8f cc = {};

    #pragma unroll
    for (int c = 0; c < NPART / 64; ++c) {   // 16 chunks
        v2f al, ac2;
        const int base = c * 64 + m * 4 + koff;
        al.x  = pl[base];     al.y  = pl[base + 1];
        ac2.x = pc[base];     ac2.y = pc[base + 1];
        cl = __builtin_amdgcn_wmma_f32_16x16x4_f32(false, al,  false, ones,
                                                   (short)0, cl, false, false);
        cc = __builtin_amdgcn_wmma_f32_16x16x4_f32(false, ac2, false, ones,
                                                   (short)0, cc, false, false);
    }

    float sl = 0.0f, sc = 0.0f;
    #pragma unroll
    for (int r = 0; r < 8; ++r) { sl += cl[r]; sc += cc[r]; }
    // lane L holds sum over M rows {0..7} (L<16) or {8..15} (L>=16) of its column
    sl += __shfl_xor(sl, 16);
    sc += __shfl_xor(sc, 16);

    if (lane == 0) out[0] = sl / sc;
}

// ---------------------------------------------------------------------------
extern "C" void kernel_launch(void* const* d_in, const int* in_sizes, int n_in,
                              void* d_out, int out_size, void* d_ws, size_t ws_size,
                              hipStream_t stream) {
    (void)in_sizes; (void)n_in; (void)out_size; (void)ws_size;

    const float* loc_pred = (const float*)d_in[0];
    // d_in[1] = conf_pred: 424 MB that never affects the output -> never read.
    const float* anchors  = (const float*)d_in[2];
    const int*   ignore   = (const int*)d_in[3];
    const float* targets  = (const float*)d_in[4];
    float*       out      = (float*)d_out;

    char* ws = (char*)d_ws;
    unsigned long long* gkeys = (unsigned long long*)ws;              // B*G*8   = 12.8 KB
    unsigned int* matchw = (unsigned int*)(ws + 16384);               // B*A*4   = 2 MB
    float* pl = (float*)(ws + 16384 + (size_t)BQ * AQ * 4);           // 1024 floats
    float* pc = pl + NPART;                                           // 1024 floats

    fgd_init_kernel<<<(BQ * GQ + 255) / 256, 256, 0, stream>>>(gkeys);
    fgd_match_kernel<<<dim3(NBQ, BQ), 256, 0, stream>>>(anchors, targets, gkeys, matchw);
    fgd_scatter_kernel<<<BQ, 32, 0, stream>>>(targets, gkeys, matchw);
    fgd_loss_kernel<<<dim3(NBQ, BQ), 256, 0, stream>>>(loc_pred, anchors, ignore,
                                                       targets, matchw, pl, pc);
    fgd_finalize_kernel<<<1, 32, 0, stream>>>(pl, pc, out);
}